// LandmarkGCN_47399259079111
// MI455X (gfx1250) — compile-verified
//
#include <hip/hip_runtime.h>
#include <hip/hip_bf16.h>

typedef __attribute__((ext_vector_type(2))) float v2f;
typedef __attribute__((ext_vector_type(8))) float v8f;

// ---------------------------------------------------------------------------
// 1) In-degree via float atomics: deg[dst[e]] += 1
// ---------------------------------------------------------------------------
__global__ void gcn_deg_kernel(const int* __restrict__ dst, float* __restrict__ deg, int E) {
    int e = blockIdx.x * blockDim.x + threadIdx.x;
    if (e < E) atomicAdd(&deg[dst[e]], 1.0f);
}

// ---------------------------------------------------------------------------
// 2) deg += 1 (self loop); dis = rsqrt(deg); dinv = 1/deg
// ---------------------------------------------------------------------------
__global__ void gcn_norm_kernel(float* __restrict__ deg, float* __restrict__ dis,
                                float* __restrict__ dinv, int N) {
    int i = blockIdx.x * blockDim.x + threadIdx.x;
    if (i < N) {
        float d = deg[i] + 1.0f;
        deg[i]  = d;
        dis[i]  = rsqrtf(d);
        dinv[i] = 1.0f / d;
    }
}

// ---------------------------------------------------------------------------
// 3) Layer-1 linear: h1[i][j] = x[i][0]*W1[0][j] + x[i][1]*W1[1][j]   (K=2)
// ---------------------------------------------------------------------------
__global__ void gcn_lin1_kernel(const float* __restrict__ x, const float* __restrict__ W1,
                                float* __restrict__ h1, int N) {
    int t = blockIdx.x * blockDim.x + threadIdx.x;   // N*32 threads
    if (t < N * 32) {
        int i = t >> 5;
        int j = t & 31;
        h1[t] = fmaf(x[2 * i], W1[j], x[2 * i + 1] * W1[32 + j]);
    }
}

// ---------------------------------------------------------------------------
// 4) Edge aggregation: agg[dst] += h[src] * (dis[src]*dis[dst])
//    One wave per edge, lanes cover feature dims (coalesced loads + atomics).
// ---------------------------------------------------------------------------
__global__ void gcn_edge_agg_kernel(const int* __restrict__ src, const int* __restrict__ dst,
                                    const float* __restrict__ dis, const float* __restrict__ h,
                                    float* __restrict__ agg, int E, int F) {
    long long t = (long long)blockIdx.x * blockDim.x + threadIdx.x;
    if (t >= (long long)E * 32) return;
    int e = (int)(t >> 5);        // whole wave shares one edge -> scalar broadcast loads
    int j = (int)(t & 31);
    int s = src[e];
    int d = dst[e];
    float nrm = dis[s] * dis[d];
    const float* hs = h + (size_t)s * F;
    float* ad = agg + (size_t)d * F;
    atomicAdd(&ad[j], hs[j] * nrm);
    if (F == 64) atomicAdd(&ad[j + 32], hs[j + 32] * nrm);
}

// ---------------------------------------------------------------------------
// 5) Fused self-loop + bias + ReLU (in place on agg)
// ---------------------------------------------------------------------------
__global__ void gcn_self_bias_relu_kernel(const float* __restrict__ h, const float* __restrict__ dinv,
                                          const float* __restrict__ b, float* __restrict__ agg,
                                          int N, int F) {
    long long t = (long long)blockIdx.x * blockDim.x + threadIdx.x;
    if (t >= (long long)N * F) return;
    int i = (int)(t / F);
    int j = (int)(t % F);
    float v = agg[t] + h[t] * dinv[i] + b[j];
    agg[t] = v > 0.0f ? v : 0.0f;
}

// ---------------------------------------------------------------------------
// 6) Layer-2 GEMM via fp32 WMMA: C[N x 64] = A[N x 32] @ B[32 x 64]
//    Block = 128 threads = 4 waves; wave w owns 16x16 output tile at col w*16.
//    K=32 accumulated as 8 x V_WMMA_F32_16X16X4_F32.
//    Epilogue: single tile-level bounds branch -> 8 straight-line stores with
//    immediate offsets (no per-row exec-mask juggling).
// ---------------------------------------------------------------------------
__global__ void __launch_bounds__(128)
gcn_gemm2_wmma_kernel(const float* __restrict__ A, const float* __restrict__ B,
                      float* __restrict__ C, int N) {
    int lane    = threadIdx.x & 31;
    int wave    = threadIdx.x >> 5;          // 0..3 -> column tile
    int rowTile = blockIdx.x;                // 16 rows per tile

    int mIdx  = lane & 15;                   // row within tile (A) / col within tile (B)
    int kHalf = (lane >> 4) << 1;            // lanes 0-15 -> K pair {0,1}; 16-31 -> {2,3}
    int row   = rowTile * 16 + mIdx;
    int col   = wave * 16 + mIdx;

    bool rowOK = row < N;
    const float* arow = A + (size_t)(rowOK ? row : 0) * 32;

    v8f acc = {};
#pragma unroll
    for (int kk = 0; kk < 32; kk += 4) {
        v2f a, b;
        float a0 = arow[kk + kHalf];
        float a1 = arow[kk + kHalf + 1];
        a.x = rowOK ? a0 : 0.0f;             // select, not branch: EXEC stays all-1s
        a.y = rowOK ? a1 : 0.0f;
        b.x = B[(kk + kHalf) * 64 + col];
        b.y = B[(kk + kHalf + 1) * 64 + col];
        // 8 args: (neg_a, A, neg_b, B, c_mod, C, reuse_a, reuse_b)
        acc = __builtin_amdgcn_wmma_f32_16x16x4_f32(
            false, a, false, b, (short)0, acc, false, false);
    }

    // C/D layout: VGPR r -> M = r (lanes 0-15) or M = r+8 (lanes 16-31); N = lane&15
    int baseM = rowTile * 16 + ((lane < 16) ? 0 : 8);
    float* crow = C + (size_t)baseM * 64 + col;
    if (rowTile * 16 + 16 <= N) {
        // full tile in range (always taken when N % 16 == 0): straight-line stores
#pragma unroll
        for (int r = 0; r < 8; ++r)
            crow[(size_t)r * 64] = acc[r];
    } else {
#pragma unroll
        for (int r = 0; r < 8; ++r)
            if (baseM + r < N) crow[(size_t)r * 64] = acc[r];
    }
}

// ---------------------------------------------------------------------------
// 7) Column mean: 1024 striped partials per column, then 64-wide atomics.
// ---------------------------------------------------------------------------
__global__ void gcn_mean_kernel(const float* __restrict__ a2, float* __restrict__ out, int N) {
    int tid  = blockIdx.x * blockDim.x + threadIdx.x;  // 65536 threads
    int col  = tid & 63;
    int part = tid >> 6;                               // 0..1023
    float s = 0.0f;
    for (int i = part; i < N; i += 1024)
        s += a2[(size_t)i * 64 + col];
    atomicAdd(&out[col], s * (1.0f / (float)N));
}

// ---------------------------------------------------------------------------
extern "C" void kernel_launch(void* const* d_in, const int* in_sizes, int n_in,
                              void* d_out, int out_size, void* d_ws, size_t ws_size,
                              hipStream_t stream) {
    const int N = in_sizes[0] / 2;   // IN_DIM = 2
    const int E = in_sizes[1] / 2;   // edge_index is [2, E]

    const float* x   = (const float*)d_in[0];
    const int*   ei  = (const int*)d_in[1];
    const float* W1  = (const float*)d_in[2];
    const float* b1  = (const float*)d_in[3];
    const float* W2  = (const float*)d_in[4];
    const float* b2  = (const float*)d_in[5];
    const int* srcI = ei;
    const int* dstI = ei + E;

    float* ws   = (float*)d_ws;
    float* deg  = ws;                       // N
    float* dis  = deg  + N;                 // N
    float* dinv = dis  + N;                 // N
    float* h1   = dinv + N;                 // N*32  (layer-1 linear)
    float* a1   = h1   + (size_t)N * 32;    // N*32  (agg -> relu'd hidden)
    float* h2   = a1   + (size_t)N * 32;    // N*64  (layer-2 linear)
    float* a2   = h2   + (size_t)N * 64;    // N*64  (agg -> relu'd output)
    float* out  = (float*)d_out;            // 64

    // Fresh zeros every call (graph-capturable).
    hipMemsetAsync(deg, 0, (size_t)N * sizeof(float), stream);
    hipMemsetAsync(a1,  0, (size_t)N * 32 * sizeof(float), stream);
    hipMemsetAsync(a2,  0, (size_t)N * 64 * sizeof(float), stream);
    hipMemsetAsync(out, 0, (size_t)out_size * sizeof(float), stream);

    const int T = 256;

    // degrees + normalization
    gcn_deg_kernel<<<(E + T - 1) / T, T, 0, stream>>>(dstI, deg, E);
    gcn_norm_kernel<<<(N + T - 1) / T, T, 0, stream>>>(deg, dis, dinv, N);

    // layer 1
    gcn_lin1_kernel<<<((long long)N * 32 + T - 1) / T, T, 0, stream>>>(x, W1, h1, N);
    gcn_edge_agg_kernel<<<(unsigned)(((long long)E * 32 + T - 1) / T), T, 0, stream>>>(
        srcI, dstI, dis, h1, a1, E, 32);
    gcn_self_bias_relu_kernel<<<(unsigned)(((long long)N * 32 + T - 1) / T), T, 0, stream>>>(
        h1, dinv, b1, a1, N, 32);

    // layer 2: WMMA GEMM (a1 is the relu'd hidden state)
    gcn_gemm2_wmma_kernel<<<(N + 15) / 16, 128, 0, stream>>>(a1, W2, h2, N);
    gcn_edge_agg_kernel<<<(unsigned)(((long long)E * 32 + T - 1) / T), T, 0, stream>>>(
        srcI, dstI, dis, h2, a2, E, 64);
    gcn_self_bias_relu_kernel<<<(unsigned)(((long long)N * 64 + T - 1) / T), T, 0, stream>>>(
        h2, dinv, b2, a2, N, 64);

    // mean over nodes
    gcn_mean_kernel<<<256, 256, 0, stream>>>(a2, out, N);
}